// TensorProductScatter_24412594111107
// MI455X (gfx1250) — compile-verified
//
#include <hip/hip_runtime.h>
#include <cstddef>

// ---------------------------------------------------------------------------
// TensorProductScatter for MI455X (gfx1250, wave32).
//
// Strategy: deterministic CSR-by-dst build (count -> scan -> place -> sort),
// then a gather kernel with ONE WAVE PER NODE, lane u = channel u (MUL==32
// matches wave32 exactly). Accumulate the node's 224 outputs in registers,
// plain coalesced stores — zero float atomics, deterministic summation order.
// Edge stream (edge_weight / edge_attr) is read once -> non-temporal loads so
// L2 (192MB) keeps x (25.6MB) and out (44.8MB) resident.
// ---------------------------------------------------------------------------

#define MUL 32

typedef float f32x4 __attribute__((ext_vector_type(4)));   // NT-loadable vec4

__global__ void zero_i32_kernel(int* __restrict__ p, int n) {
    int i = blockIdx.x * blockDim.x + threadIdx.x;
    if (i < n) p[i] = 0;
}

__global__ void count_kernel(const int* __restrict__ dst, int* __restrict__ counts, int E) {
    int e = blockIdx.x * blockDim.x + threadIdx.x;
    if (e < E) atomicAdd(&counts[dst[e]], 1);
}

// Single-workgroup looped exclusive scan (n=50000 -> 49 chunks of 1024).
__global__ void scan_kernel(const int* __restrict__ counts, int* __restrict__ offsets,
                            int n, int E) {
    __shared__ int s[1024];
    __shared__ int carry_s;
    if (threadIdx.x == 0) carry_s = 0;
    __syncthreads();
    int nchunk = (n + 1023) / 1024;
    for (int c = 0; c < nchunk; ++c) {
        int i = c * 1024 + threadIdx.x;
        int v = (i < n) ? counts[i] : 0;
        s[threadIdx.x] = v;
        __syncthreads();
        for (int off = 1; off < 1024; off <<= 1) {
            int t = (threadIdx.x >= off) ? s[threadIdx.x - off] : 0;
            __syncthreads();
            s[threadIdx.x] += t;
            __syncthreads();
        }
        int incl = s[threadIdx.x];
        if (i < n) offsets[i] = carry_s + incl - v;   // exclusive prefix
        __syncthreads();                               // all reads of carry_s done
        if (threadIdx.x == 0) carry_s += s[1023];
        __syncthreads();
    }
    if (threadIdx.x == 0) offsets[n] = E;
}

__global__ void place_kernel(const int* __restrict__ dst, const int* __restrict__ offsets,
                             int* __restrict__ cursor, int* __restrict__ edge_ids, int E) {
    int e = blockIdx.x * blockDim.x + threadIdx.x;
    if (e < E) {
        int d = dst[e];
        int slot = offsets[d] + atomicAdd(&cursor[d], 1);
        edge_ids[slot] = e;
    }
}

// Deterministic per-node edge order: insertion-sort each node's short list
// (Poisson mean 16) so float accumulation order is identical every run.
__global__ void sort_kernel(const int* __restrict__ offsets, int* __restrict__ edge_ids, int n) {
    int v = blockIdx.x * blockDim.x + threadIdx.x;
    if (v >= n) return;
    int b = offsets[v], t = offsets[v + 1];
    for (int i = b + 1; i < t; ++i) {
        int key = edge_ids[i];
        int j = i - 1;
        while (j >= b && edge_ids[j] > key) { edge_ids[j + 1] = edge_ids[j]; --j; }
        edge_ids[j + 1] = key;
    }
}

// One wave per node; lane u handles channel u (wave32 == MUL).
__global__ void gather_kernel(const float* __restrict__ x,
                              const float* __restrict__ attr,
                              const float* __restrict__ w,
                              const int* __restrict__ src,
                              const int* __restrict__ offsets,
                              const int* __restrict__ edge_ids,
                              float* __restrict__ out, int n_nodes) {
    const float INV_S2 = 0.70710678118654752f;
    const float INV_S3 = 0.57735026918962576f;
    int gid  = blockIdx.x * blockDim.x + threadIdx.x;
    int node = gid >> 5;            // wave id
    int lane = threadIdx.x & 31;
    if (node >= n_nodes) return;    // whole waves exit together

    int b = offsets[node];
    int t = offsets[node + 1];

    float acc0 = 0.f;
    float a10 = 0.f, a11 = 0.f, a12 = 0.f;
    float a20 = 0.f, a21 = 0.f, a22 = 0.f;

    for (int j = b; j < t; ++j) {
        int e = __builtin_amdgcn_readfirstlane(edge_ids[j]);
        // software prefetch: next edge's weight row (gfx1250 global_prefetch_b8)
        if (j + 1 < t) {
            int en = __builtin_amdgcn_readfirstlane(edge_ids[j + 1]);
            __builtin_prefetch(w + (size_t)en * (5 * MUL), 0, 0);
        }
        int s = __builtin_amdgcn_readfirstlane(src[e]);

        // per-edge attrs (uniform across the wave), streamed once -> NT b128
        const f32x4* ap = reinterpret_cast<const f32x4*>(attr + (size_t)e * 4);
        f32x4 A  = __builtin_nontemporal_load(ap);
        float a0 = A.x, v0 = A.y, v1 = A.z, v2 = A.w;

        // per-edge weights: 5 coalesced 128B NT loads
        const float* we = w + (size_t)e * (5 * MUL);
        float w0 = __builtin_nontemporal_load(we + 0 * MUL + lane);
        float w1 = __builtin_nontemporal_load(we + 1 * MUL + lane);
        float w2 = __builtin_nontemporal_load(we + 2 * MUL + lane);
        float w3 = __builtin_nontemporal_load(we + 3 * MUL + lane);
        float w4 = __builtin_nontemporal_load(we + 4 * MUL + lane);

        // source node features (L2-resident, reused -> regular temporal)
        const float* xs = x + (size_t)s * (4 * MUL);
        float x0  = xs[lane];
        const float* x1p = xs + MUL + 3 * lane;
        float x10 = x1p[0], x11 = x1p[1], x12 = x1p[2];

        float dot = x10 * v0 + x11 * v1 + x12 * v2;
        acc0 += INV_S2 * (w0 * x0 * a0 + w3 * (INV_S3 * dot));

        float t1 = INV_S2 * w1 * x0;
        float t2 = INV_S2 * w2 * a0;
        a10 += t1 * v0 + t2 * x10;
        a11 += t1 * v1 + t2 * x11;
        a12 += t1 * v2 + t2 * x12;

        float c0 = x11 * v2 - x12 * v1;
        float c1 = x12 * v0 - x10 * v2;
        float c2 = x10 * v1 - x11 * v0;
        float t4 = INV_S2 * w4;
        a20 += t4 * c0;
        a21 += t4 * c1;
        a22 += t4 * c2;
    }

    float* o = out + (size_t)node * (7 * MUL);
    o[lane]                   = acc0;        // out0
    o[MUL     + 3 * lane + 0] = a10;         // out1 (u,i) -> 32 + 3u + i
    o[MUL     + 3 * lane + 1] = a11;
    o[MUL     + 3 * lane + 2] = a12;
    o[4 * MUL + 3 * lane + 0] = a20;         // out2 -> 128 + 3u + i
    o[4 * MUL + 3 * lane + 1] = a21;
    o[4 * MUL + 3 * lane + 2] = a22;
}

// ---------------------------------------------------------------------------
// TDM compile probe (NOT launched): verifies the 6-arg clang-23
// __builtin_amdgcn_tensor_load_to_lds form assembles for gfx1250 and puts
// tensor_load_to_lds / s_wait_tensorcnt into the object. Null descriptor
// (count=0). If this compiles, a later round can use gather-mode TDM to stage
// per-node weight rows into LDS.
// ---------------------------------------------------------------------------
typedef unsigned int uint32x4 __attribute__((ext_vector_type(4)));
typedef int          int32x8  __attribute__((ext_vector_type(8)));
typedef int          int32x4  __attribute__((ext_vector_type(4)));

__global__ void tdm_probe_kernel() {
    uint32x4 g0 = {0, 0, 0, 0};
    int32x8  g1 = {0, 0, 0, 0, 0, 0, 0, 0};
    int32x4  g2 = {0, 0, 0, 0};
    int32x4  g3 = {0, 0, 0, 0};
    int32x8  g4 = {0, 0, 0, 0, 0, 0, 0, 0};
    __builtin_amdgcn_tensor_load_to_lds(g0, g1, g2, g3, g4, 0);
    __builtin_amdgcn_s_wait_tensorcnt(0);
}

extern "C" void kernel_launch(void* const* d_in, const int* in_sizes, int n_in,
                              void* d_out, int out_size, void* d_ws, size_t ws_size,
                              hipStream_t stream) {
    (void)n_in; (void)out_size; (void)ws_size;

    const float* x    = (const float*)d_in[0];
    const float* attr = (const float*)d_in[1];
    const float* w    = (const float*)d_in[2];
    const int*   dst  = (const int*)d_in[3];
    const int*   src  = (const int*)d_in[4];

    const int N = in_sizes[0] / (4 * MUL);  // 50000
    const int E = in_sizes[3];              // 800000

    int* ws       = (int*)d_ws;
    int* counts   = ws;                        // N
    int* cursor   = ws + N;                    // N
    int* offsets  = ws + 2 * (size_t)N;        // N+1
    int* edge_ids = ws + 3 * (size_t)N + 1;    // E

    const int T = 256;

    // 1) zero counts + cursor (contiguous 2N ints)
    zero_i32_kernel<<<(2 * N + T - 1) / T, T, 0, stream>>>(counts, 2 * N);
    // 2) histogram of dst
    count_kernel<<<(E + T - 1) / T, T, 0, stream>>>(dst, counts, E);
    // 3) exclusive scan -> CSR offsets
    scan_kernel<<<1, 1024, 0, stream>>>(counts, offsets, N, E);
    // 4) place edge ids into CSR slots
    place_kernel<<<(E + T - 1) / T, T, 0, stream>>>(dst, offsets, cursor, edge_ids, E);
    // 5) sort each node's list -> deterministic accumulation order
    sort_kernel<<<(N + T - 1) / T, T, 0, stream>>>(offsets, edge_ids, N);
    // 6) fused tensor-product + gather-sum: one wave32 per node
    int gather_threads = N * 32;
    gather_kernel<<<(gather_threads + T - 1) / T, T, 0, stream>>>(
        x, attr, w, src, offsets, edge_ids, (float*)d_out, N);
}